// EmoStyle_75273596830036
// MI455X (gfx1250) — compile-verified
//
#include <hip/hip_runtime.h>
#include <stdint.h>

#define B_SZ     64
#define N_TOK    256
#define IN_DIM   8
#define IMG_DIM  1152
#define T_NUM    192
#define EMB      3072
#define HALF_EMB 1536
#define CODEBOOK 512
#define LN_EPS   1e-5f

typedef __attribute__((ext_vector_type(2))) float v2f;
typedef __attribute__((ext_vector_type(4))) float v4f;
typedef __attribute__((ext_vector_type(8))) float v8f;

// ---- CDNA5 async global<->LDS copy (ASYNCcnt path), per CDNA5 ISA ----
__device__ __forceinline__ void async_g2l_b128(uint32_t lds_off, const void* gaddr) {
    // copies 16 bytes from the global address gaddr to the LDS offset lds_off
    asm volatile("global_load_async_to_lds_b128 %0, %1, off"
                 :: "v"(lds_off), "v"(gaddr) : "memory");
}
__device__ __forceinline__ void async_l2g_b128(void* gaddr, uint32_t lds_off) {
    // copies 16 bytes from the LDS offset lds_off to the global address gaddr
    asm volatile("global_store_async_from_lds_b128 %0, %1, off"
                 :: "v"(gaddr), "v"(lds_off) : "memory");
}
__device__ __forceinline__ void wait_asynccnt0() {
    asm volatile("s_wait_asynccnt 0x0" ::: "memory");
}

// ---------------- Kernel 1: img_pooled = mean_n(image_tokens) ----------------
// grid 64, block 288 (one float4 per thread). 75 MB read-once -> NT loads.
__global__ void k_pool(const float* __restrict__ img, float* __restrict__ pooled) {
    const int b  = blockIdx.x;
    const int d4 = threadIdx.x;
    const v4f* src = (const v4f*)(img + (size_t)b * N_TOK * IMG_DIM);
    v4f acc = {};
    for (int n = 0; n < N_TOK; ++n)
        acc += __builtin_nontemporal_load(src + n * (IMG_DIM / 4) + d4);
    acc *= (1.f / (float)N_TOK);
    ((v4f*)(pooled + (size_t)b * IMG_DIM))[d4] = acc;
}

// ---------------- Kernel 2: q = l2norm(layernorm(concat(emo@We+be, pool@Wi+bi)))
// grid 64 (one block per batch row), block 512
__global__ void k_query(const float* __restrict__ emo,
                        const float* __restrict__ pooled,
                        const float* __restrict__ W_emo, const float* __restrict__ b_emo,
                        const float* __restrict__ W_img, const float* __restrict__ b_img,
                        float* __restrict__ q) {
    const int b = blockIdx.x;
    const int t = threadIdx.x;

    __shared__ float pool_s[IMG_DIM];
    __shared__ float emo_s[IN_DIM];
    __shared__ float q_s[EMB];
    __shared__ float red_s[512];
    __shared__ float red_q[512];
    __shared__ float bcast[2];

    for (int i = t; i < IMG_DIM; i += 512) pool_s[i] = pooled[(size_t)b * IMG_DIM + i];
    if (t < IN_DIM) emo_s[t] = emo[(size_t)b * IN_DIM + t];
    __syncthreads();

    // ---- emo half: 3 columns per thread (c = t + 512*r), K = 8 ----
    #pragma unroll
    for (int r = 0; r < 3; ++r) {
        const int c = t + 512 * r;
        float a = b_emo[c];
        #pragma unroll
        for (int k = 0; k < IN_DIM; ++k) a += emo_s[k] * W_emo[k * HALF_EMB + c];
        q_s[c] = a;
    }

    // ---- img half: 3 columns per thread, K = 1152, coalesced W_img rows ----
    float a0 = b_img[t], a1 = b_img[t + 512], a2 = b_img[t + 1024];
    for (int k = 0; k < IMG_DIM; ++k) {
        const float p = pool_s[k];
        const float* wrow = W_img + (size_t)k * HALF_EMB;
        a0 += p * wrow[t];
        a1 += p * wrow[t + 512];
        a2 += p * wrow[t + 1024];
    }
    q_s[HALF_EMB + t]        = a0;
    q_s[HALF_EMB + t + 512]  = a1;
    q_s[HALF_EMB + t + 1024] = a2;
    __syncthreads();

    // ---- layernorm stats over 3072 ----
    float ls = 0.f, lq = 0.f;
    for (int i = t; i < EMB; i += 512) { float v = q_s[i]; ls += v; lq += v * v; }
    red_s[t] = ls; red_q[t] = lq;
    __syncthreads();
    for (int s = 256; s > 0; s >>= 1) {
        if (t < s) { red_s[t] += red_s[t + s]; red_q[t] += red_q[t + s]; }
        __syncthreads();
    }
    if (t == 0) {
        const float mean = red_s[0] / (float)EMB;
        float var = red_q[0] / (float)EMB - mean * mean;
        var = fmaxf(var, 0.f);
        bcast[0] = mean;
        bcast[1] = rsqrtf(var + LN_EPS);
    }
    __syncthreads();
    const float mean = bcast[0], rstd = bcast[1];

    // ---- normalize, then L2 norm of the normalized vector ----
    float sq = 0.f;
    for (int i = t; i < EMB; i += 512) {
        float v = (q_s[i] - mean) * rstd;
        q_s[i] = v;
        sq += v * v;
    }
    red_s[t] = sq;
    __syncthreads();
    for (int s = 256; s > 0; s >>= 1) {
        if (t < s) red_s[t] += red_s[t + s];
        __syncthreads();
    }
    if (t == 0) bcast[0] = 1.f / fmaxf(sqrtf(red_s[0]), 1e-6f);
    __syncthreads();
    const float invn = bcast[0];
    for (int i = t; i < EMB; i += 512) q[(size_t)b * EMB + i] = q_s[i] * invn;
}

// ---------------- Kernel 3: code_repr = mean_t(style_dict); scale = 1/max(||.||,1e-6)
// grid 512 (one block per code), block 256 — the 1.2 GB streaming pass (NT loads,
// so q/code_repr tiles reused by the WMMA kernel stay resident in the 192 MB L2)
__global__ void k_code(const float* __restrict__ sd,
                       float* __restrict__ code_repr, float* __restrict__ scale) {
    const int c = blockIdx.x;
    const int t = threadIdx.x;
    const v4f* src = (const v4f*)(sd + (size_t)c * T_NUM * EMB);
    v4f s0 = {}, s1 = {}, s2 = {};
    for (int tok = 0; tok < T_NUM; ++tok) {
        const v4f* row = src + tok * (EMB / 4);
        s0 += __builtin_nontemporal_load(row + t);
        s1 += __builtin_nontemporal_load(row + t + 256);
        s2 += __builtin_nontemporal_load(row + t + 512);
    }
    const float inv = 1.f / (float)T_NUM;
    s0 *= inv; s1 *= inv; s2 *= inv;
    v4f* dst = (v4f*)(code_repr + (size_t)c * EMB);
    dst[t] = s0; dst[t + 256] = s1; dst[t + 512] = s2;

    float sq = s0.x*s0.x + s0.y*s0.y + s0.z*s0.z + s0.w*s0.w
             + s1.x*s1.x + s1.y*s1.y + s1.z*s1.z + s1.w*s1.w
             + s2.x*s2.x + s2.y*s2.y + s2.z*s2.z + s2.w*s2.w;
    __shared__ float red[256];
    red[t] = sq;
    __syncthreads();
    for (int s = 128; s > 0; s >>= 1) {
        if (t < s) red[t] += red[t + s];
        __syncthreads();
    }
    if (t == 0) scale[c] = 1.f / fmaxf(sqrtf(red[0]), 1e-6f);   // TAU == 1
}

// ---------------- Kernel 4: logits = (q @ code_reprT) * scale  via f32 WMMA
// grid 32 (N tiles of 16), block 128 = 4 waves (M tiles of 16); M=64,N=512,K=3072
__global__ void k_logits(const float* __restrict__ q,
                         const float* __restrict__ code_repr,
                         const float* __restrict__ scale,
                         const unsigned char* __restrict__ value_init,
                         float* __restrict__ logits) {
    const int lane = threadIdx.x & 31;
    const int wave = threadIdx.x >> 5;
    const int m0 = wave * 16;
    const int n0 = blockIdx.x * 16;
    const int row16 = lane & 15;       // M (for A) / N (for B,C,D)
    const int koff  = (lane >> 4) * 2; // K pair selected by lane half

    const float* arow = q         + (size_t)(m0 + row16) * EMB;  // A: M x K
    const float* brow = code_repr + (size_t)(n0 + row16) * EMB;  // B: K x N = code_repr[n][k]

    v8f acc = {};
    for (int k = 0; k < EMB; k += 4) {
        v2f a, b;
        a.x = arow[k + koff]; a.y = arow[k + koff + 1];
        b.x = brow[k + koff]; b.y = brow[k + koff + 1];
        acc = __builtin_amdgcn_wmma_f32_16x16x4_f32(
            false, a, false, b, (short)0, acc, false, false);
    }

    const int n = n0 + row16;
    const float sc = scale[n];
    const bool keep = value_init[n] != 0;
    #pragma unroll
    for (int i = 0; i < 8; ++i) {
        const int m = m0 + (lane >> 4) * 8 + i;
        float v = acc[i] * sc;
        if (!keep) v = -1e9f;
        logits[(size_t)m * CODEBOOK + n] = v;
    }
}

// ---------------- Kernel 5: argmax per row (first-index tie-break like jnp.argmax)
// grid 64, block 256
__global__ void k_argmax(const float* __restrict__ logits,
                         int* __restrict__ idx, float* __restrict__ idx_f) {
    const int b = blockIdx.x;
    const int t = threadIdx.x;
    __shared__ float vmax[256];
    __shared__ int   vidx[256];
    float best = -3.402823466e+38f;
    int   bi   = 0;
    for (int c = t; c < CODEBOOK; c += 256) {
        float v = logits[(size_t)b * CODEBOOK + c];
        if (v > best || (v == best && c < bi)) { best = v; bi = c; }
    }
    vmax[t] = best; vidx[t] = bi;
    __syncthreads();
    for (int s = 128; s > 0; s >>= 1) {
        if (t < s) {
            if (vmax[t + s] > vmax[t] ||
                (vmax[t + s] == vmax[t] && vidx[t + s] < vidx[t])) {
                vmax[t] = vmax[t + s]; vidx[t] = vidx[t + s];
            }
        }
        __syncthreads();
    }
    if (t == 0) { idx[b] = vidx[0]; idx_f[b] = (float)vidx[0]; }
}

// ---------------- Kernel 6: out[b] = style_dict[idx[b]]  (one-hot ST gather)
// grid (192, 64), block 256. 151 MB copy routed through the CDNA5 async
// global<->LDS pipe: each thread moves 3x16B g->LDS, waits ASYNCcnt, 3x16B LDS->g.
// Per-wave self-consistent (each wave stores only bytes it loaded), so only the
// per-wave s_wait_asynccnt is needed; s_endpgm drains the async stores.
__global__ void k_gather(const float* __restrict__ sd,
                         const int* __restrict__ idx, float* __restrict__ out) {
    __shared__ __align__(16) float buf[EMB];   // 12 KB staging tile
    const int tok = blockIdx.x;
    const int b   = blockIdx.y;
    const int c   = idx[b];
    const char* src = (const char*)(sd + ((size_t)c * T_NUM + tok) * EMB);
    char*       dst = (char*)(out + ((size_t)b * T_NUM + tok) * EMB);
    const uint32_t lds0 = (uint32_t)(uintptr_t)buf;   // low 32 bits of flat LDS addr = LDS offset
    const int t = threadIdx.x;
    #pragma unroll
    for (int i = 0; i < 3; ++i) {
        const uint32_t off = (uint32_t)(t * 16 + i * 4096);
        async_g2l_b128(lds0 + off, src + off);
    }
    wait_asynccnt0();
    #pragma unroll
    for (int i = 0; i < 3; ++i) {
        const uint32_t off = (uint32_t)(t * 16 + i * 4096);
        async_l2g_b128(dst + off, lds0 + off);
    }
}

extern "C" void kernel_launch(void* const* d_in, const int* in_sizes, int n_in,
                              void* d_out, int out_size, void* d_ws, size_t ws_size,
                              hipStream_t stream) {
    (void)in_sizes; (void)n_in; (void)out_size; (void)ws_size;
    const float* emo   = (const float*)d_in[0];
    const float* img   = (const float*)d_in[1];
    const float* W_emo = (const float*)d_in[2];
    const float* b_emo = (const float*)d_in[3];
    const float* W_img = (const float*)d_in[4];
    const float* b_img = (const float*)d_in[5];
    const float* sd    = (const float*)d_in[6];
    const unsigned char* vinit = (const unsigned char*)d_in[7];
    float* out = (float*)d_out;

    // workspace layout (floats)
    float* ws        = (float*)d_ws;
    float* pooled    = ws;                            // 64*1152
    float* q         = pooled + B_SZ * IMG_DIM;       // 64*3072
    float* code_repr = q + B_SZ * EMB;                // 512*3072
    float* scale     = code_repr + CODEBOOK * EMB;    // 512
    float* logits    = scale + CODEBOOK;              // 64*512
    int*   idx       = (int*)(logits + B_SZ * CODEBOOK); // 64

    float* out_idx_f = out + (size_t)B_SZ * T_NUM * EMB; // argmax appended to d_out

    k_pool  <<<B_SZ, IMG_DIM / 4, 0, stream>>>(img, pooled);
    k_query <<<B_SZ, 512, 0, stream>>>(emo, pooled, W_emo, b_emo, W_img, b_img, q);
    k_code  <<<CODEBOOK, 256, 0, stream>>>(sd, code_repr, scale);
    k_logits<<<CODEBOOK / 16, 128, 0, stream>>>(q, code_repr, scale, vinit, logits);
    k_argmax<<<B_SZ, 256, 0, stream>>>(logits, idx, out_idx_f);
    k_gather<<<dim3(T_NUM, B_SZ), 256, 0, stream>>>(sd, idx, out);
}